// BiLSTM_CRF_42339787604517
// MI455X (gfx1250) — compile-verified
//
#include <hip/hip_runtime.h>

// CRF NLL for B=1024, T=512, K=32 on gfx1250 (wave32, WMMA).
//
// Kernel 1 (gold+len): one wave per sequence, lanes stride over T.
// Kernel 2 (partition): one wave per 16-sequence tile. The logsumexp DP step
//   alpha'[n] = LSE_p(alpha[p]+trans[n,p]) + feat[n]
// runs in exp-space as a 16x32 @ 32x32 GEMM on V_WMMA_F32_16X16X4_F32
// (2 N-tiles x 8 K-chunks = 16 WMMA/step). The whole exponential pipeline is
// base-2 (raw v_exp_f32 / v_log_f32 + FMA folding) so no libm denorm-fixup
// sequences land on the serial critical path:
//   E2[n,p] = 2^(trans[n,p]*log2e)            (setup, INVALID -> exactly 0)
//   U2[p]   = 2^((alpha[p]-m)*log2e)          (1 fma + 1 v_exp per element)
//   V       = U2 @ E2^T  == e^(alpha+trans-m) (WMMA)
//   alpha'  = fma(v_log(V), ln2, m + feat)    (1 v_log + 1 add + 1 fma)

#define KTAGS     32
#define START_TAG 30
#define END_TAG   31
#define INVALID_F (-10000.0f)
#define LOG2E_F   1.44269504088896340736f
#define LN2_F     0.69314718055994530942f

typedef __attribute__((ext_vector_type(2))) float v2f;
typedef __attribute__((ext_vector_type(8))) float v8f;

static __device__ __forceinline__ float exp2f_raw(float x) {
  return __builtin_amdgcn_exp2f(x);  // v_exp_f32 (base 2)
}
static __device__ __forceinline__ float log2f_raw(float x) {
  return __builtin_amdgcn_logf(x);   // v_log_f32 (base 2)
}

// ---------------------------------------------------------------------------
// Kernel 1: gold score + lengths. One wave (32 lanes) per batch row.
// ---------------------------------------------------------------------------
__global__ __launch_bounds__(256) void crf_gold_len_kernel(
    const float* __restrict__ feats, const float* __restrict__ trans,
    const int* __restrict__ tags, const int* __restrict__ masks,
    float* __restrict__ gold, int* __restrict__ lens, int T)
{
  const int lane = threadIdx.x & 31;
  const int wid  = threadIdx.x >> 5;
  const int b    = blockIdx.x * (blockDim.x >> 5) + wid;

  const int*   tg = tags  + (size_t)b * T;
  const int*   mk = masks + (size_t)b * T;
  const float* fb = feats + (size_t)b * T * KTAGS;

  float acc = 0.0f;
  int   len = 0;
  for (int t = lane; t < T; t += 32) {
    int m    = mk[t];
    int tagt = tg[t];
    len += m;
    if (m) acc += fb[(size_t)t * KTAGS + tagt];                 // emit * mask
    // transition into position t (i>=1): trans[tags[t], tags[t-1]] * mask[t-1]
    if (t > 0) {
      if (mk[t - 1]) acc += trans[tagt * KTAGS + tg[t - 1]];
    }
  }
  if (lane == 0)                                                 // i==0 term
    acc += trans[tg[0] * KTAGS + START_TAG];
  if (lane == ((T - 1) & 31)) {                                  // i==T term
    if (mk[T - 1]) acc += trans[END_TAG * KTAGS + tg[T - 1]];
  }
  // wave32 butterfly reduce
  for (int k = 16; k >= 1; k >>= 1) {
    acc += __shfl_xor(acc, k, 32);
    len += __shfl_xor(len, k, 32);
  }
  if (lane == 0) { gold[b] = acc; lens[b] = len; }
}

// ---------------------------------------------------------------------------
// Kernel 2: log-partition via exp-space WMMA DP + fused (Z - gold) output.
// One wave per 16-batch tile.
//
// Layouts (wave32, per CDNA5 ISA):
//   A  (16x4 f32):  vgpr j, lane L:  M = L%16, K = 4c + 2*(L>=16) + j
//   B  ( 4x16 f32): vgpr j, lane L:  K = 4c + 2*(L>=16) + j (mirrors A), N = L%16
//   C/D(16x16 f32): vgpr v, lane L:  M = v + 8*(L>=16),  N = L%16
// ---------------------------------------------------------------------------
__global__ __launch_bounds__(32) void crf_forward_wmma_kernel(
    const float* __restrict__ feats, const float* __restrict__ trans,
    const float* __restrict__ gold, const int* __restrict__ lens,
    float* __restrict__ out, int T)
{
  const int lane = threadIdx.x;   // 0..31
  const int lo   = lane & 15;
  const int hi   = lane >> 4;     // 0 or 1
  const int b0   = blockIdx.x * 16;

  __shared__ float s_alpha[16 * KTAGS];  // [m][k] transpose staging (2 KB)
  __shared__ float s_c[16];              // per-row rescale constants

  // --- B operand: E2[p][n] = 2^(trans[n][p]*log2e); INVALID -> exactly 0.
  v2f bmat[2][8];
  #pragma unroll
  for (int tile = 0; tile < 2; ++tile) {
    const int n = lo + 16 * tile;
    #pragma unroll
    for (int c = 0; c < 8; ++c) {
      const int p = 4 * c + 2 * hi;
      bmat[tile][c].x = exp2f_raw(trans[n * KTAGS + p] * LOG2E_F);
      bmat[tile][c].y = exp2f_raw(trans[n * KTAGS + p + 1] * LOG2E_F);
    }
  }
  // trans[END][k] in A-domain layout (for the terminal LSE)
  v2f tEA[8];
  #pragma unroll
  for (int c = 0; c < 8; ++c) {
    const int p = 4 * c + 2 * hi;
    tEA[c].x = trans[END_TAG * KTAGS + p];
    tEA[c].y = trans[END_TAG * KTAGS + p + 1];
  }

  // --- lengths for the 8 rows this lane-half owns (m = v + 8*hi)
  int lenv[8];
  #pragma unroll
  for (int v = 0; v < 8; ++v) lenv[v] = lens[b0 + v + 8 * hi];
  int tmax = lenv[0];
  #pragma unroll
  for (int v = 1; v < 8; ++v) tmax = max(tmax, lenv[v]);
  tmax = max(tmax, __shfl_xor(tmax, 16, 32));
  tmax = __builtin_amdgcn_readfirstlane(tmax);

  // --- alpha init (D layout): START row = 0, rest = INVALID
  float a0[8], a1[8];
  #pragma unroll
  for (int v = 0; v < 8; ++v) {
    a0[v] = INVALID_F;                                   // n = lo (< 16)
    a1[v] = (lo == (START_TAG - 16)) ? 0.0f : INVALID_F; // n = lo + 16
  }

  const size_t rstride = (size_t)T * KTAGS;  // elements per batch row
  const float* fp = feats + (size_t)(b0 + 8 * hi) * rstride + lo;

  for (int t = 0; t < tmax; ++t) {
    // feats in D layout: f0 -> n=lo, f1 -> n=lo+16, row m = v + 8*hi
    float f0[8], f1[8];
    #pragma unroll
    for (int v = 0; v < 8; ++v) {
      f0[v] = fp[v * rstride];
      f1[v] = fp[v * rstride + 16];
    }
    if ((t & 7) == 0) {   // stage next 8 timesteps toward L2/WGP$
      #pragma unroll
      for (int v = 0; v < 8; ++v)
        __builtin_prefetch(fp + v * rstride + 8 * KTAGS, 0, 1);
    }

    // D-layout alpha -> LDS [m][k]
    #pragma unroll
    for (int v = 0; v < 8; ++v) {
      s_alpha[(v + 8 * hi) * KTAGS + lo]      = a0[v];
      s_alpha[(v + 8 * hi) * KTAGS + lo + 16] = a1[v];
    }
    __builtin_amdgcn_wave_barrier();

    // transposed read in A layout: row m = lo, k = 4c + 2*hi + {0,1}
    v2f aT[8];
    #pragma unroll
    for (int c = 0; c < 8; ++c)
      aT[c] = *(const v2f*)&s_alpha[lo * KTAGS + 4 * c + 2 * hi];

    // per-row max (15 in-lane + cross-half combine)
    float mrow = fmaxf(aT[0].x, aT[0].y);
    #pragma unroll
    for (int c = 1; c < 8; ++c) mrow = fmaxf(mrow, fmaxf(aT[c].x, aT[c].y));
    mrow = fmaxf(mrow, __shfl_xor(mrow, 16, 32));
    const float negm2 = -mrow * LOG2E_F;

    // U2 = 2^((alpha - m)*log2e): WMMA A operand, already in A layout
    v2f u[8];
    #pragma unroll
    for (int c = 0; c < 8; ++c) {
      u[c].x = exp2f_raw(fmaf(aT[c].x, LOG2E_F, negm2));
      u[c].y = exp2f_raw(fmaf(aT[c].y, LOG2E_F, negm2));
    }

    // rescale constants back to D domain via LDS
    s_c[lo] = mrow;
    __builtin_amdgcn_wave_barrier();
    float cD[8];
    #pragma unroll
    for (int v = 0; v < 8; ++v) cD[v] = s_c[v + 8 * hi];

    // V[16b x 32n] = U2[16b x 32p] @ E2[32p x 32n]: 2 N-tiles x 8 K-chunks
    v8f acc0 = {0.f, 0.f, 0.f, 0.f, 0.f, 0.f, 0.f, 0.f};
    v8f acc1 = acc0;
    #pragma unroll
    for (int c = 0; c < 8; ++c)
      acc0 = __builtin_amdgcn_wmma_f32_16x16x4_f32(
          false, u[c], false, bmat[0][c], (short)0, acc0, false, false);
    #pragma unroll
    for (int c = 0; c < 8; ++c)
      acc1 = __builtin_amdgcn_wmma_f32_16x16x4_f32(
          false, u[c], false, bmat[1][c], (short)0, acc1, false, false);

    // alpha' = ln(V) + m + feat = fma(log2(V), ln2, m+feat); freeze t>=len[m]
    #pragma unroll
    for (int v = 0; v < 8; ++v) {
      const bool live = t < lenv[v];
      float n0 = fmaf(log2f_raw(acc0[v]), LN2_F, cD[v] + f0[v]);
      float n1 = fmaf(log2f_raw(acc1[v]), LN2_F, cD[v] + f1[v]);
      a0[v] = live ? n0 : a0[v];
      a1[v] = live ? n1 : a1[v];
    }
    fp += KTAGS;
  }

  // --- terminal: Z[m] = LSE_n(alpha[m,n] + trans[END,n]); transpose once more
  #pragma unroll
  for (int v = 0; v < 8; ++v) {
    s_alpha[(v + 8 * hi) * KTAGS + lo]      = a0[v];
    s_alpha[(v + 8 * hi) * KTAGS + lo + 16] = a1[v];
  }
  __builtin_amdgcn_wave_barrier();

  v2f zT[8];
  #pragma unroll
  for (int c = 0; c < 8; ++c) {
    v2f aT = *(const v2f*)&s_alpha[lo * KTAGS + 4 * c + 2 * hi];
    zT[c].x = aT.x + tEA[c].x;
    zT[c].y = aT.y + tEA[c].y;
  }
  float mz = fmaxf(zT[0].x, zT[0].y);
  #pragma unroll
  for (int c = 1; c < 8; ++c) mz = fmaxf(mz, fmaxf(zT[c].x, zT[c].y));
  mz = fmaxf(mz, __shfl_xor(mz, 16, 32));
  const float negmz2 = -mz * LOG2E_F;
  float sz = 0.0f;
  #pragma unroll
  for (int c = 0; c < 8; ++c)
    sz += exp2f_raw(fmaf(zT[c].x, LOG2E_F, negmz2)) +
          exp2f_raw(fmaf(zT[c].y, LOG2E_F, negmz2));
  sz += __shfl_xor(sz, 16, 32);
  const float Z = fmaf(log2f_raw(sz), LN2_F, mz);

  if (hi == 0)  // lanes 0..15 own rows m = 0..15 after the transpose
    out[b0 + lo] = Z - gold[b0 + lo];
}

// ---------------------------------------------------------------------------
extern "C" void kernel_launch(void* const* d_in, const int* in_sizes, int n_in,
                              void* d_out, int out_size, void* d_ws, size_t ws_size,
                              hipStream_t stream) {
  const float* feats = (const float*)d_in[0];   // [B,T,K] f32
  const float* trans = (const float*)d_in[1];   // [K,K]   f32
  const int*   tags  = (const int*)d_in[2];     // [B,T]   i32
  const int*   masks = (const int*)d_in[3];     // [B,T]   i32
  float*       out   = (float*)d_out;           // [B]     f32

  const int B = out_size;                       // 1024
  const int T = in_sizes[2] / B;                // 512
  (void)n_in; (void)ws_size;

  float* gold = (float*)d_ws;
  int*   lens = (int*)((char*)d_ws + (size_t)B * sizeof(float));

  const int wavesPerBlock = 8;
  crf_gold_len_kernel<<<B / wavesPerBlock, 32 * wavesPerBlock, 0, stream>>>(
      feats, trans, tags, masks, gold, lens, T);
  crf_forward_wmma_kernel<<<B / 16, 32, 0, stream>>>(
      feats, trans, gold, lens, out, T);
}